// CFMToyModel_65017214927270
// MI455X (gfx1250) — compile-verified
//
#include <hip/hip_runtime.h>
#include <math.h>

#define NPART 2048
#define EPSF  1e-6f
#define HSTR  36            // hidden K stride: 32 channels + ones/bias chunk (4)

typedef __attribute__((ext_vector_type(2))) float v2f;
typedef __attribute__((ext_vector_type(8))) float v8f;

__device__ __forceinline__ v8f wmma4(v2f a, v2f b, v8f c) {
  // D = A(16x4,f32) * B(4x16,f32) + C(16x16,f32), exact f32 semantics
  return __builtin_amdgcn_wmma_f32_16x16x4_f32(false, a, false, b, (short)0, c,
                                               false, false);
}
__device__ __forceinline__ v2f ld2(const float* p) { return *(const v2f*)p; }

// ---------------------------------------------------------------------------
// Kernel 1: base velocity MLP  (2048 x (3 -> 64 -> 64 -> 2)), ~18 MFLOP, VALU.
// Writes base velocity into d_out; pairwise kernel accumulates on top.
// ---------------------------------------------------------------------------
__global__ __launch_bounds__(256) void base_mlp_kernel(
    const float* __restrict__ x,
    const float* __restrict__ W1, const float* __restrict__ b1,
    const float* __restrict__ W2, const float* __restrict__ b2,
    const float* __restrict__ W3, const float* __restrict__ b3,
    float* __restrict__ out)
{
  int i = blockIdx.x * blockDim.x + threadIdx.x;   // grid covers N exactly
  float in0 = x[i*3+0], in1 = x[i*3+1], in2 = x[i*3+2];

  float h1[64];
#pragma unroll
  for (int c = 0; c < 64; ++c) {
    float v = b1[c] + in0 * W1[c] + in1 * W1[64+c] + in2 * W1[128+c];
    h1[c] = v > 0.f ? v : 0.f;
  }

  float o0 = b3[0], o1 = b3[1];
  for (int c = 0; c < 64; ++c) {
    float v = b2[c];
#pragma unroll
    for (int k = 0; k < 64; ++k) v += h1[k] * W2[k*64 + c];
    v = v > 0.f ? v : 0.f;
    o0 += v * W3[c*2 + 0];
    o1 += v * W3[c*2 + 1];
  }
  out[i*2+0] = o0;
  out[i*2+1] = o1;
}

// ---------------------------------------------------------------------------
// Kernel 2: pairwise-force MLP via V_WMMA_F32_16X16X4_F32 (wave32).
// Block = 256 threads = 8 waves = 2 particles x 4 j-segments (512 j's each).
// Per 16-pair chunk:
//   L1: feat(16x8,  ones@k=3 carries b1) x W1t            -> 2 wmma x 2 tiles
//   L2: h1 (16x36,  ones@k=32 carries b2) x W2t           -> 9 wmma x 2 tiles
//   L3: h2 (16x36,  ones@k=32 carries b3) x W3t(N pad 16) -> 9 wmma
// All C accumulators start as inline 0 (biases folded into the matmuls).
// Weights staged TRANSPOSED ([n][k]) in LDS so each B fragment is one
// aligned ds_load_b64; A fragments likewise contiguous.
// Fragment layouts (32-bit, wave32):
//   A 16x4 : lane<16 row=lane k={0,1};  lane>=16 row=lane-16 k={2,3}
//   B 4x16 : lane<16 col=lane k={0,1};  lane>=16 col=lane-16 k={2,3}
//   C 16x16: col = lane&15, row = vgpr + 8*(lane>>4)
// ---------------------------------------------------------------------------
__global__ __launch_bounds__(256) void pair_forces_kernel(
    const float* __restrict__ x,
    const float* __restrict__ gW1, const float* __restrict__ gb1,
    const float* __restrict__ gW2, const float* __restrict__ gb2,
    const float* __restrict__ gW3, const float* __restrict__ gb3,
    float* __restrict__ out)
{
  __shared__ float sW1t[32*8];      // [n][k]; k padded 5->8; b1 folded at k=3
  __shared__ float sW2t[32*HSTR];   // [n][k]; k<32 = W2, k==32 = b2, else 0
  __shared__ float sW3t[16*HSTR];   // [n][k]; n padded 2->16; k==32 = b3
  __shared__ float sFeat[8][16*8];
  __shared__ float sH[8][16*HSTR];  // reused for h1 then h2; cols 32..35 fixed
  __shared__ float sRed[8][2];

  const int tid = threadIdx.x;

  // ---- stage transposed / padded / bias-folded weights ----
  for (int t = tid; t < 32*8; t += 256) {
    int n = t >> 3, k = t & 7;
    float v = (k < 5) ? gW1[k*32 + n] : 0.f;
    if (k == 3) v += gb1[n];              // ones feature carries bias
    sW1t[t] = v;
  }
  for (int t = tid; t < 32*HSTR; t += 256) {
    int n = t / HSTR, k = t % HSTR;
    float v = 0.f;
    if (k < 32)       v = gW2[k*32 + n];
    else if (k == 32) v = gb2[n];
    sW2t[t] = v;
  }
  for (int t = tid; t < 16*HSTR; t += 256) {
    int n = t / HSTR, k = t % HSTR;
    float v = 0.f;
    if (n < 2) {
      if (k < 32)       v = gW3[k*2 + n];
      else if (k == 32) v = gb3[n];
    }
    sW3t[t] = v;
  }

  const int wave = tid >> 5;
  const int lane = tid & 31;
  const int iLoc = wave >> 2;          // 0..1 : particle within block
  const int seg  = wave & 3;           // 0..3 : j segment
  const int i    = blockIdx.x * 2 + iLoc;
  const int r    = lane & 15;          // fragment row/col within tile
  const int hi   = lane >> 4;          // half-wave select
  const int kb   = hi * 2;             // A/B k-offset within 4-wide chunk
  const int mh   = hi * 8;             // C row offset

  float* feat = sFeat[wave];
  float* hbuf = sH[wave];

  // one-time constants in per-wave slabs (never overwritten in the loop)
  if (lane < 16) {
    feat[r*8 + 3] = 1.f;                                  // ones feature
    feat[r*8 + 5] = 0.f; feat[r*8 + 6] = 0.f; feat[r*8 + 7] = 0.f;
    hbuf[r*HSTR + 32] = 1.f;                              // ones channel
    hbuf[r*HSTR + 33] = 0.f; hbuf[r*HSTR + 34] = 0.f; hbuf[r*HSTR + 35] = 0.f;
  }
  __syncthreads();

  const float ixp = x[i*3+0];
  const float iyp = x[i*3+1];

  float accL = 0.f;                    // partial column sum for channel n=r
  const int jBeg = seg * (NPART/4);

  for (int j0 = jBeg; j0 < jBeg + NPART/4; j0 += 16) {
    // ---- features for pairs (i, j0+r): [dxn_x, dxn_y, dist, 1, t_j] ----
    int jdx = j0 + r;
    float jx = x[jdx*3+0], jy = x[jdx*3+1], jt = x[jdx*3+2];
    float dxv = ixp - jx, dyv = iyp - jy;
    float sq   = dxv*dxv + dyv*dyv;
    float dist = sq > 0.f ? sqrtf(sq) : 0.f;
    float inv  = 1.f / (dist + EPSF);
    if (lane < 16) {
      float* fr = &feat[r*8];
      fr[0] = dxv * inv; fr[1] = dyv * inv; fr[2] = dist; fr[4] = jt;
    }
    __builtin_amdgcn_wave_barrier();

    // ---- layer 1: (16x8) x (8x32), bias folded, relu -> hbuf ----
    v2f a0 = ld2(&feat[r*8 + kb]);
    v2f a1 = ld2(&feat[r*8 + 4 + kb]);
    v8f h1acc[2];
#pragma unroll
    for (int t = 0; t < 2; ++t) {
      v8f acc = {};
      acc = wmma4(a0, ld2(&sW1t[(16*t + r)*8 + kb]),     acc);
      acc = wmma4(a1, ld2(&sW1t[(16*t + r)*8 + 4 + kb]), acc);
      h1acc[t] = acc;
    }
#pragma unroll
    for (int t = 0; t < 2; ++t)
#pragma unroll
      for (int v = 0; v < 8; ++v)
        hbuf[(mh+v)*HSTR + 16*t + r] = h1acc[t][v] > 0.f ? h1acc[t][v] : 0.f;
    __builtin_amdgcn_wave_barrier();

    // ---- layer 2: (16x36) x (36x32), bias via ones channel, relu ----
    v2f a2[9];
#pragma unroll
    for (int c = 0; c < 9; ++c) a2[c] = ld2(&hbuf[r*HSTR + 4*c + kb]);
    v8f acc2[2];
#pragma unroll
    for (int t = 0; t < 2; ++t) {
      v8f acc = {};
#pragma unroll
      for (int c = 0; c < 9; ++c)
        acc = wmma4(a2[c], ld2(&sW2t[(16*t + r)*HSTR + 4*c + kb]), acc);
      acc2[t] = acc;
    }
    __builtin_amdgcn_wave_barrier();
#pragma unroll
    for (int t = 0; t < 2; ++t)
#pragma unroll
      for (int v = 0; v < 8; ++v)
        hbuf[(mh+v)*HSTR + 16*t + r] = acc2[t][v] > 0.f ? acc2[t][v] : 0.f;
    __builtin_amdgcn_wave_barrier();

    // ---- layer 3: (16x36) x (36x16 padded), bias via ones channel ----
    v8f acc3 = {};
#pragma unroll
    for (int c = 0; c < 9; ++c) {
      v2f a = ld2(&hbuf[r*HSTR + 4*c + kb]);
      v2f b = ld2(&sW3t[r*HSTR + 4*c + kb]);
      acc3 = wmma4(a, b, acc3);
    }

    // ---- masked sum over this chunk's 16 rows (exclude j == i) ----
#pragma unroll
    for (int v = 0; v < 8; ++v) {
      int jj = j0 + mh + v;
      accL += (jj == i) ? 0.f : acc3[v];
    }
  }

  // combine half-column partials within wave (rows 0..7 + rows 8..15)
  float other  = __shfl_xor(accL, 16, 32);
  float colsum = accL + other;
  if (hi == 0 && r < 2) sRed[wave][r] = colsum;
  __syncthreads();

  // combine the 4 j-segments per particle; single writer per output element
  if (tid < 4) {
    int il = tid >> 1, n = tid & 1;
    float s = sRed[il*4+0][n] + sRed[il*4+1][n] +
              sRed[il*4+2][n] + sRed[il*4+3][n];
    int ii = blockIdx.x * 2 + il;
    out[ii*2 + n] += s / (float)(NPART - 1);   // base written by kernel 1
  }
}

// ---------------------------------------------------------------------------
extern "C" void kernel_launch(void* const* d_in, const int* in_sizes, int n_in,
                              void* d_out, int out_size, void* d_ws, size_t ws_size,
                              hipStream_t stream) {
  const float* x   = (const float*)d_in[0];
  const float* bW1 = (const float*)d_in[1];
  const float* bb1 = (const float*)d_in[2];
  const float* bW2 = (const float*)d_in[3];
  const float* bb2 = (const float*)d_in[4];
  const float* bW3 = (const float*)d_in[5];
  const float* bb3 = (const float*)d_in[6];
  const float* iW1 = (const float*)d_in[7];
  const float* ib1 = (const float*)d_in[8];
  const float* iW2 = (const float*)d_in[9];
  const float* ib2 = (const float*)d_in[10];
  const float* iW3 = (const float*)d_in[11];
  const float* ib3 = (const float*)d_in[12];
  float* out = (float*)d_out;

  // 1) base velocity -> d_out
  base_mlp_kernel<<<NPART / 256, 256, 0, stream>>>(x, bW1, bb1, bW2, bb2,
                                                   bW3, bb3, out);
  // 2) pairwise forces accumulated on top (8192 waves: 2 i x 4 j-segs / block)
  pair_forces_kernel<<<NPART / 2, 256, 0, stream>>>(x, iW1, ib1, iW2, ib2,
                                                    iW3, ib3, out);
}